// AdvancedNoncommutativeManifold_14285061226505
// MI455X (gfx1250) — compile-verified
//
#include <hip/hip_runtime.h>
#include <stdint.h>

#define BASE 2048
#define CDIM 512
#define QDIM 256
#define TOT  (BASE + CDIM + QDIM)   // 2816
#define CS   BASE                   // 2048
#define CE   (BASE + CDIM)          // 2560
#define THETA_C      1e-25
#define THETA_COSMIC 1e-27

// ---------- helpers ----------

__device__ __forceinline__ void async_copy_b128(unsigned lds_off, const void* gaddr) {
    // CDNA5 async global->LDS copy (16B per lane), tracked by ASYNCcnt.
    asm volatile("global_load_async_to_lds_b128 %0, %1, off"
                 :
                 : "v"(lds_off), "v"((unsigned long long)gaddr)
                 : "memory");
}

__device__ __forceinline__ void wait_async_all() {
    asm volatile("s_wait_asynccnt 0" ::: "memory");
}

__device__ __forceinline__ unsigned lds_addr_of(const void* p) {
    // low 32 bits of a generic shared-aperture address == LDS byte offset
    return (unsigned)(uintptr_t)p;
}

__device__ __forceinline__ double block_reduce256(double v, double* red) {
    int t = threadIdx.x;
    red[t] = v; __syncthreads();
    for (int off = 128; off > 0; off >>= 1) {
        if (t < off) red[t] += red[t + off];
        __syncthreads();
    }
    double r = red[0]; __syncthreads();
    return r;
}

__device__ __forceinline__ double zeta_re(int r, double sr, double si) {
    // diagonal zeta term, hermitized => Re(exp(-s ln n))
    double ln = log((double)(r + 1));
    double lr = -sr * ln, li = -si * ln;
    bool small_s = (fabs(sr) < 20.0) && (fabs(si) < 200.0);
    if (small_s || lr > -50.0) return exp(lr) * cos(li);
    return 1e-50;
}

// ---------- kernel 1: norms ----------
// ws[0] = 1/||v||, ws[1] = ||v/||v|||| (vn), ws[2] = ||x/||x|||| (cnorm)
__global__ __launch_bounds__(256)
void norms_kernel(const double2* __restrict__ v, const double2* __restrict__ cr,
                  double* __restrict__ ws) {
    __shared__ double red[256];
    int t = threadIdx.x;

    double s = 0.0;
    for (int i = t; i < CDIM; i += 256) { double2 a = v[i]; s += a.x*a.x + a.y*a.y; }
    double inv = 1.0 / sqrt(block_reduce256(s, red));

    s = 0.0;
    for (int i = t; i < CDIM; i += 256) {
        double2 a = v[i]; double xr = a.x*inv, xi = a.y*inv; s += xr*xr + xi*xi;
    }
    double vn = sqrt(block_reduce256(s, red));

    s = 0.0;
    for (int i = t; i < BASE; i += 256) { double2 a = cr[i]; s += a.x*a.x + a.y*a.y; }
    double invc = 1.0 / sqrt(block_reduce256(s, red));

    s = 0.0;
    for (int i = t; i < BASE; i += 256) {
        double2 a = cr[i]; double xr = a.x*invc, xi = a.y*invc; s += xr*xr + xi*xi;
    }
    double cnorm = sqrt(block_reduce256(s, red));

    if (t == 0) { ws[0] = inv; ws[1] = vn; ws[2] = cnorm; }
}

// ---------- kernel 2: base block [0:2048)^2 ----------
// One workgroup per tile-pair (bi<=bj): reads tiles K[bi,bj] and K[bj,bi] once,
// produces both hermitized output tiles. Async global->LDS copies.
__global__ __launch_bounds__(256)
void base_block_kernel(const double2* __restrict__ K,
                       const double2* __restrict__ gs,   // [4,8,8]
                       const double2* __restrict__ gr,   // [4,16,16]
                       const double* __restrict__ srp, const double* __restrict__ sip,
                       const double* __restrict__ ws, double2* __restrict__ H) {
    int bj = blockIdx.x, bi = blockIdx.y;
    if (bi > bj) return;

    __shared__ double2 tA[16][17];
    __shared__ double2 tB[16][17];

    int tid = threadIdx.x;
    int tx = tid & 15, ty = tid >> 4;
    int r0 = bi * 16, c0 = bj * 16;

    async_copy_b128(lds_addr_of(&tA[ty][tx]), &K[(r0 + ty) * BASE + (c0 + tx)]);
    async_copy_b128(lds_addr_of(&tB[ty][tx]), &K[(c0 + ty) * BASE + (r0 + tx)]);
    wait_async_all();
    __syncthreads();

    double th = THETA_COSMIC * ws[2];         // theta_cosmic * cnorm
    double sr = srp[0], si = sip[0];

    // upper tile (bi,bj): element (r,c) = 0.5*th*(K[r,c] + conj(K[c,r]))
    double2 A  = tA[ty][tx];
    double2 Bt = tB[tx][ty];
    double2 val = make_double2(0.5 * th * (A.x + Bt.x), 0.5 * th * (A.y - Bt.y));

    int r = r0 + ty, c = c0 + tx;
    if (r == c) val.x += zeta_re(r, sr, si) + 1e-20;

    if (r < 16 && c < 16) {  // gamma corner (only tile (0,0))
        double2 acc = make_double2(0.0, 0.0);
        if (r < 8 && c < 8) {
            #pragma unroll
            for (int k = 0; k < 4; k++) {
                double sc = (double)(k + 1) * (THETA_C / 10.0);
                double2 a = gs[k * 64 + r * 8 + c];
                double2 b = gs[k * 64 + c * 8 + r];
                acc.x += sc * 0.5 * (a.x + b.x);
                acc.y += sc * 0.5 * (a.y - b.y);
            }
        }
        #pragma unroll
        for (int k = 0; k < 4; k++) {
            double sc = (double)(k + 5) * (THETA_C / 10.0);
            double2 a = gr[k * 256 + r * 16 + c];
            double2 b = gr[k * 256 + c * 16 + r];
            acc.x += sc * 0.5 * (a.x + b.x);
            acc.y += sc * 0.5 * (a.y - b.y);   // antihermitian => cancels exactly
        }
        val.x += acc.x; val.y += acc.y;
    }
    H[r * TOT + c] = val;

    if (bi != bj) {  // lower mirror tile (bj,bi)
        double2 B  = tB[ty][tx];
        double2 At = tA[tx][ty];
        double2 v2 = make_double2(0.5 * th * (B.x + At.x), 0.5 * th * (B.y - At.y));
        H[(c0 + ty) * TOT + (r0 + tx)] = v2;
    }
}

// ---------- kernel 3: coupling blocks [:2048,2048:2560) and mirror ----------
// Hermitized closed form: H[i, CS+j] = C[j,i]*vn ; H[CS+j, i] = conj(C[j,i])*vn
__global__ __launch_bounds__(256)
void coupling_kernel(const double2* __restrict__ C,   // [CDIM, BASE] row-major
                     const double* __restrict__ ws, double2* __restrict__ H) {
    int it = blockIdx.x;   // 0..127 over BASE
    int jt = blockIdx.y;   // 0..31  over CDIM
    __shared__ double2 tC[16][17];

    int tid = threadIdx.x;
    int tx = tid & 15, ty = tid >> 4;
    int i0 = it * 16, j0 = jt * 16;

    async_copy_b128(lds_addr_of(&tC[ty][tx]), &C[(j0 + ty) * BASE + (i0 + tx)]);
    wait_async_all();
    __syncthreads();

    double vn = ws[1];

    double2 u = tC[tx][ty];                       // C[j0+tx][i0+ty]
    H[(i0 + ty) * TOT + (CS + j0 + tx)] = make_double2(u.x * vn, u.y * vn);

    double2 w = tC[ty][tx];                       // C[j0+ty][i0+tx]
    H[(CS + j0 + ty) * TOT + (i0 + tx)] = make_double2(w.x * vn, -w.y * vn);
}

// ---------- kernel 4: consciousness block [2048:2560)^2 ----------
__global__ __launch_bounds__(256)
void cons_kernel(const double2* __restrict__ v, const double* __restrict__ ws,
                 double2* __restrict__ H) {
    int idx = blockIdx.x * 256 + threadIdx.x;
    if (idx >= CDIM * CDIM) return;
    int a = idx >> 9, b = idx & 511;
    double inv = ws[0];
    double2 va = v[a], vb = v[b];
    double ar = va.x * inv, ai = va.y * inv;
    double br = vb.x * inv, bi = vb.y * inv;
    double re = (ar * br + ai * bi) * THETA_C;    // v_a * conj(v_b)
    double im = (ai * br - ar * bi) * THETA_C;
    if (a == b) re += 1e-20;
    H[(CS + a) * TOT + (CS + b)] = make_double2(re, im);
}

// ---------- kernel 5: zero strips + quantum diagonal ----------
__global__ __launch_bounds__(256)
void strip_kernel(double2* __restrict__ H, int r0, int c0, int nr, int nc,
                  const double* __restrict__ srp, const double* __restrict__ sip) {
    long idx = (long)blockIdx.x * 256 + threadIdx.x;
    if (idx >= (long)nr * nc) return;
    int r = r0 + (int)(idx / nc);
    int c = c0 + (int)(idx % nc);
    double2 val = make_double2(0.0, 0.0);
    if (r == c && r >= CE) {
        double sr = srp[0], si = sip[0];
        double smag = sqrt(sr * sr + si * si);
        double entropy = -smag * log(smag + 1e-10) * (1.0 + 0.1 * sin(si / 10.0));
        val.x = entropy / (double)(r - CE + 1) + 1e-20;
    }
    H[r * TOT + c] = val;
}

// ---------- launch ----------
extern "C" void kernel_launch(void* const* d_in, const int* in_sizes, int n_in,
                              void* d_out, int out_size, void* d_ws, size_t ws_size,
                              hipStream_t stream) {
    const double*  sr = (const double*)d_in[0];
    const double*  si = (const double*)d_in[1];
    const double2* v  = (const double2*)d_in[2];   // consciousness_vector [512]
    const double2* cr = (const double2*)d_in[3];   // cosmic_ray_data [2048]
    const double2* C  = (const double2*)d_in[4];   // coupling_cr [512,2048]
    const double2* K  = (const double2*)d_in[5];   // cosmic_coupling [2048,2048]
    const double2* gs = (const double2*)d_in[6];   // gamma_small [4,8,8]
    const double2* gr = (const double2*)d_in[7];   // gamma_rand  [4,16,16]
    double2* H  = (double2*)d_out;
    double*  ws = (double*)d_ws;

    norms_kernel<<<1, 256, 0, stream>>>(v, cr, ws);

    dim3 gb(BASE / 16, BASE / 16);                 // 128x128, bi>bj exits early
    base_block_kernel<<<gb, 256, 0, stream>>>(K, gs, gr, sr, si, ws, H);

    dim3 gc(BASE / 16, CDIM / 16);                 // 128x32
    coupling_kernel<<<gc, 256, 0, stream>>>(C, ws, H);

    cons_kernel<<<(CDIM * CDIM + 255) / 256, 256, 0, stream>>>(v, ws, H);

    // right strip: all rows, cols [CE, TOT) — includes quantum diagonal
    strip_kernel<<<(TOT * QDIM + 255) / 256, 256, 0, stream>>>(H, 0, CE, TOT, QDIM, sr, si);
    // bottom strip: rows [CE, TOT), cols [0, CE)
    strip_kernel<<<(QDIM * CE + 255) / 256, 256, 0, stream>>>(H, CE, 0, QDIM, CE, sr, si);
}